// FISM_54760833024108
// MI455X (gfx1250) — compile-verified
//
#include <hip/hip_runtime.h>

typedef __attribute__((ext_vector_type(2))) float v2f;
typedef __attribute__((ext_vector_type(4))) float v4f;
typedef __attribute__((ext_vector_type(8))) float v8f;

namespace {
constexpr int kB        = 16384;   // batch rows
constexpr int kK        = 50;      // neighbors per row
constexpr int kNNeg     = 20;      // negatives per row
constexpr int kD        = 128;     // embedding dim
constexpr int kTgt      = 21;      // 1 positive + 20 negatives
constexpr int kQRows    = 32;      // padded target rows (2 WMMA M-groups)
constexpr int kRowPad   = 132;     // floats per LDS row: 128 + 4 pad -> conflict-free ds_load_b64
constexpr int kRowsPerB = 2;       // one wave32 per batch row, 2 rows per block
}

__global__ __launch_bounds__(32 * kRowsPerB)
void fism_fwd(const int* __restrict__ I, const int* __restrict__ U,
              const int* __restrict__ Ineg, const int* __restrict__ IU,
              const int* __restrict__ NU, const int* __restrict__ Ind,
              const float* __restrict__ P, const float* __restrict__ Q,
              const float* __restrict__ bu, const float* __restrict__ bi,
              float* __restrict__ r, float* __restrict__ rneg)
{
    __shared__ float sq[kRowsPerB][kQRows][kRowPad];
    __shared__ float sp[kRowsPerB][kRowPad];
    __shared__ float sdot[kRowsPerB][kQRows];

    const int lane = threadIdx.x & 31;
    const int wv   = threadIdx.x >> 5;
    const int b    = blockIdx.x * kRowsPerB + wv;   // grid sized exactly: b < kB

    const int ipos = I[b];

    // Prefetch the q-rows we will need later, overlapping with the neighbor sum.
    if (lane < kTgt) {
        const int qi = (lane == 0) ? ipos : Ineg[b * kNNeg + (lane - 1)];
        __builtin_prefetch(Q + (size_t)qi * kD, 0, 3);   // global_prefetch_b8
    }

    // ---- 1) gather + sum 50 neighbor embeddings (coalesced b128 per row) ----
    const long base = (long)b * kK;
    v4f acc = {};
    #pragma unroll 5
    for (int t = 0; t < kK; ++t) {
        const int idx = IU[base + t];                       // uniform -> s_load
        acc += ((const v4f*)(P + (size_t)idx * kD))[lane];  // global_load_b128
    }

    // ---- 2) exclude self, normalize ----
    const float find  = (float)Ind[b];
    const v4f   pself = ((const v4f*)(P + (size_t)ipos * kD))[lane];
    acc -= pself * find;
    const float nf = (float)NU[b];
    const float dv = fmaxf(nf - find, 1.0f);
    acc /= dv;                                   // p_ctx_avg, lane holds dims 4l..4l+3
    *(v4f*)&sp[wv][lane * 4] = acc;

    // ---- 3) stage 21 q-vectors (target 0 = positive) into padded LDS rows ----
    #pragma unroll
    for (int t = 0; t < kTgt; ++t) {
        const int qi = (t == 0) ? ipos : Ineg[b * kNNeg + (t - 1)];
        *(v4f*)&sq[wv][t][lane * 4] = ((const v4f*)(Q + (size_t)qi * kD))[lane];
    }
    #pragma unroll
    for (int t = kTgt; t < kQRows; ++t) {
        v4f z = {};
        *(v4f*)&sq[wv][t][lane * 4] = z;
    }

    __syncthreads();

    // ---- 4) dots via V_WMMA_F32_16X16X4_F32 (full f32 precision) ----
    // A layout (16x4 f32): lanes 0-15 hold {K=0,K=1}, lanes 16-31 hold {K=2,K=3} for row M=lane%16.
    // B (4x16): broadcast the same p-chunk pair across all 16 columns.
    const int m  = lane & 15;
    const int kb = (lane >> 4) << 1;             // 0 or 2

    #pragma unroll
    for (int g = 0; g < 2; ++g) {
        v8f c = {};
        const float* qrow = &sq[wv][g * 16 + m][0];
        const float* prow = &sp[wv][0];
        #pragma unroll
        for (int ch = 0; ch < kD / 4; ++ch) {    // 32 K-chunks of 4 dims
            const int off = ch * 4 + kb;
            v2f a  = *(const v2f*)(qrow + off);  // ds_load_b64, bank-conflict-free (pad=4)
            v2f bb = *(const v2f*)(prow + off);  // same-address broadcast
            c = __builtin_amdgcn_wmma_f32_16x16x4_f32(
                    false, a, false, bb, (short)0, c, false, false);
        }
        // C layout: VGPR v -> M=v (lanes 0-15) / M=v+8 (lanes 16-31); all columns identical.
        if (lane == 0) {
            #pragma unroll
            for (int v = 0; v < 8; ++v) sdot[wv][g * 16 + v] = c[v];
        } else if (lane == 16) {
            #pragma unroll
            for (int v = 0; v < 8; ++v) sdot[wv][g * 16 + 8 + v] = c[v];
        }
    }

    __syncthreads();

    // ---- 5) biases + outputs ----
    const float ub = bu[U[b]];
    if (lane == 0)
        r[b] = ub + bi[ipos] + sdot[wv][0];
    if (lane < kNNeg) {
        const int ni = Ineg[b * kNNeg + lane];
        rneg[b * kNNeg + lane] = ub + bi[ni] + sdot[wv][1 + lane];
    }
}

extern "C" void kernel_launch(void* const* d_in, const int* in_sizes, int n_in,
                              void* d_out, int out_size, void* d_ws, size_t ws_size,
                              hipStream_t stream) {
    const int*   I    = (const int*)d_in[0];
    const int*   U    = (const int*)d_in[1];
    const int*   Ineg = (const int*)d_in[2];
    const int*   IU   = (const int*)d_in[3];
    const int*   NU   = (const int*)d_in[4];
    const int*   Ind  = (const int*)d_in[5];
    const float* P    = (const float*)d_in[6];
    const float* Q    = (const float*)d_in[7];
    const float* bu   = (const float*)d_in[8];
    const float* bi   = (const float*)d_in[9];

    float* r    = (float*)d_out;        // [B]
    float* rneg = r + kB;               // [B, NNEG] flattened

    dim3 grid(kB / kRowsPerB);
    dim3 block(32 * kRowsPerB);
    fism_fwd<<<grid, block, 0, stream>>>(I, U, Ineg, IU, NU, Ind, P, Q, bu, bi, r, rneg);
}